// ConvTemporalGraphical_3178275799573
// MI455X (gfx1250) — compile-verified
//
#include <hip/hip_runtime.h>
#include <hip/hip_bf16.h>
#include <cstddef>

// Shapes from the reference
#define NB   32
#define CIN  4
#define COUT 64
#define TT   6
#define VV   512
#define MROWS (COUT * TT)          // 384 GEMM rows per batch
#define OUT_ELEMS ((size_t)NB * COUT * TT * VV)   // 6,291,456 (first tuple output)

typedef __attribute__((ext_vector_type(2))) float v2f;
typedef __attribute__((ext_vector_type(8))) float v8f;

// ---------------------------------------------------------------------------
// Kernel 1: xc = conv_w @ x + conv_b   (1x1 conv over channels)
//           x1[n,v,o] = sum_t xc * l2_w[t] + l2_b  (fused)
//           e_i[n,v] = x1 . l1_w[0:64],  e_j[n,v] = x1 . l1_w[64:128]
// One block per (n,v); 64 threads = one output channel each.
// ---------------------------------------------------------------------------
__global__ void conv_pool_e_kernel(const float* __restrict__ x,
                                   const float* __restrict__ conv_w,
                                   const float* __restrict__ conv_b,
                                   const float* __restrict__ l2_w,
                                   const float* __restrict__ l2_b,
                                   const float* __restrict__ l1_w,
                                   float* __restrict__ xc,
                                   float* __restrict__ e_i,
                                   float* __restrict__ e_j) {
    const int nv = blockIdx.x;          // 0 .. N*V-1
    const int n  = nv >> 9;             // / 512
    const int v  = nv & (VV - 1);
    const int o  = threadIdx.x;         // 0 .. 63

    float w[CIN];
#pragma unroll
    for (int c = 0; c < CIN; ++c) w[c] = conv_w[o * CIN + c];
    const float bias = conv_b[o];

    float x1 = 0.0f;
#pragma unroll
    for (int t = 0; t < TT; ++t) {
        float s = bias;
#pragma unroll
        for (int c = 0; c < CIN; ++c)
            s += w[c] * x[(((size_t)n * CIN + c) * TT + t) * VV + v];
        xc[(((size_t)n * COUT + o) * TT + t) * VV + v] = s;
        x1 += s * l2_w[t];
    }
    x1 += l2_b[0];

    __shared__ float sei[COUT];
    __shared__ float sej[COUT];
    sei[o] = x1 * l1_w[o];
    sej[o] = x1 * l1_w[COUT + o];
    __syncthreads();
#pragma unroll
    for (int off = 32; off > 0; off >>= 1) {
        if (o < off) { sei[o] += sei[o + off]; sej[o] += sej[o + off]; }
        __syncthreads();
    }
    if (o == 0) { e_i[nv] = sei[0]; e_j[nv] = sej[0]; }
}

// ---------------------------------------------------------------------------
// Kernel 2: scores[n,i,j] = leakyrelu(e_i + e_j + l1_b), masked by A[n,7,i,j],
//           row softmax over j with NaN->0 (all-masked row => p row = 0).
// One block (256 threads) per row (n,i); 2 columns per thread.
// ---------------------------------------------------------------------------
__global__ void attn_softmax_kernel(const int* __restrict__ A,
                                    const float* __restrict__ e_i,
                                    const float* __restrict__ e_j,
                                    const float* __restrict__ l1_b,
                                    float* __restrict__ p) {
    const int i   = blockIdx.x;     // 0 .. 511
    const int n   = blockIdx.y;     // 0 .. 31
    const int tid = threadIdx.x;    // 0 .. 255

    const float ei = e_i[n * VV + i];
    const float b  = l1_b[0];
    const int* mrow = A + ((size_t)(n * 8 + 7) * VV + i) * VV;

    float sc[2];
    int   valid[2];
    float lmax = -INFINITY;
#pragma unroll
    for (int u = 0; u < 2; ++u) {
        const int j = tid + u * 256;
        float s = ei + e_j[n * VV + j] + b;
        s = (s >= 0.0f) ? s : 0.2f * s;          // leaky relu, slope 0.2
        valid[u] = (mrow[j] != 0);
        sc[u] = valid[u] ? s : -INFINITY;
        lmax = fmaxf(lmax, sc[u]);
    }

    __shared__ float red[256];
    red[tid] = lmax;
    __syncthreads();
#pragma unroll
    for (int off = 128; off > 0; off >>= 1) {
        if (tid < off) red[tid] = fmaxf(red[tid], red[tid + off]);
        __syncthreads();
    }
    const float rowmax = red[0];
    __syncthreads();

    float le[2];
    float lsum = 0.0f;
#pragma unroll
    for (int u = 0; u < 2; ++u) {
        le[u] = valid[u] ? __expf(sc[u] - rowmax) : 0.0f;
        lsum += le[u];
    }
    red[tid] = lsum;
    __syncthreads();
#pragma unroll
    for (int off = 128; off > 0; off >>= 1) {
        if (tid < off) red[tid] += red[tid + off];
        __syncthreads();
    }
    const float inv = (red[0] > 0.0f) ? (1.0f / red[0]) : 0.0f;  // NaN->0 path

    float* prow = p + ((size_t)n * VV + i) * VV;
#pragma unroll
    for (int u = 0; u < 2; ++u) prow[tid + u * 256] = le[u] * inv;
}

// ---------------------------------------------------------------------------
// Kernel 3: out[n] = xc[n] (384x512) @ p[n] (512x512), fp32 WMMA 16x16x4.
// Block tile 128(M) x 64(N); 8 waves, each wave owns a 16x64 strip (4 accs).
//
// B chunk staged in LDS in K-PAIR layout: for even k, B[k][col] and B[k+1][col]
// are adjacent dwords, so a lane's whole B fragment is ONE ds_load_b64 landing
// in an aligned VGPR pair (no v_mov shuffles before the WMMA). Pair-row stride
// is 160 dwords (128 data + 32 pad) so lanes 16-31 (reading k2+1) hit the
// other 32 LDS banks than lanes 0-15.
//
// Fragment layouts per CDNA5 ISA 7.12.2:
//   A 16x4 f32:  v0 = {lanes 0-15: K=0 | lanes 16-31: K=2}
//                v1 = {lanes 0-15: K=1 | lanes 16-31: K=3}, M = lane%16
//   B 4x16 f32:  v0 = {lanes 0-15: row K=0 | lanes 16-31: row K=2}
//                v1 = {lanes 0-15: row K=1 | lanes 16-31: row K=3}, N = lane%16
//   C/D 16x16:   VGPR r = {lanes 0-15: M=r | lanes 16-31: M=r+8}, N = lane%16
// ---------------------------------------------------------------------------
#define TM 128
#define TN 64
#define TKC 16
#define PAIR_STRIDE 160   // dwords per k-pair row: 2*64 data + 32 pad

__global__ void gemm_xc_p_kernel(const float* __restrict__ xc,
                                 const float* __restrict__ p,
                                 float* __restrict__ out) {
    const int n     = blockIdx.z;
    const int tileN = blockIdx.x * TN;
    const int tileM = blockIdx.y * TM;

    const float* Abase = xc  + (size_t)n * MROWS * VV;
    const float* Bbase = p   + (size_t)n * VV * VV;
    float*       Obase = out + (size_t)n * MROWS * VV;

    __shared__ __align__(16) float Bs[(TKC / 2) * PAIR_STRIDE];   // 5 KB

    const int tid  = threadIdx.x;           // 0..255
    const int wid  = tid >> 5;              // 0..7  -> M subtile
    const int lane = tid & 31;
    const int lhalf = lane >> 4;            // 0: lanes 0-15, 1: lanes 16-31
    const int l15   = lane & 15;

    const int mrow = tileM + wid * 16 + l15;      // A row for this lane
    const int kOff = lhalf * 2;                   // K offset {0,2} per half-wave

    v8f acc[4] = {};                              // 4 n-subtiles of 16x16

    for (int k0 = 0; k0 < VV; k0 += TKC) {
        __syncthreads();
        // Stage B chunk (rows k0..k0+15, cols tileN..tileN+63) in k-pair layout
#pragma unroll
        for (int u = 0; u < (TKC * TN) / 256; ++u) {
            const int idx = tid + u * 256;   // 0..1023
            const int kk  = idx >> 6;        // 0..15
            const int nn  = idx & 63;
            Bs[(kk >> 1) * PAIR_STRIDE + nn * 2 + (kk & 1)] =
                Bbase[(size_t)(k0 + kk) * VV + tileN + nn];
        }
        // Prefetch next B chunk into cache while we compute this one
        if (k0 + TKC < VV)
            __builtin_prefetch(Bbase + (size_t)(k0 + TKC) * VV + tileN + ((tid & 15) << 2), 0, 1);
        __syncthreads();

#pragma unroll
        for (int kk = 0; kk < TKC; kk += 4) {
            // A fragment: one 64-bit global load per lane
            const float* aptr = Abase + (size_t)mrow * VV + (k0 + kk + kOff);
            const v2f a = *(const v2f*)aptr;
            const int k2 = (kk + kOff) >> 1;       // pair-row index
#pragma unroll
            for (int t = 0; t < 4; ++t) {
                const int col = t * 16 + l15;
                // B fragment: one 64-bit LDS load per lane (aligned pair)
                const v2f bfrag = *(const v2f*)&Bs[k2 * PAIR_STRIDE + col * 2];
                acc[t] = __builtin_amdgcn_wmma_f32_16x16x4_f32(
                    /*neg_a=*/false, a, /*neg_b=*/false, bfrag,
                    /*c_mod=*/(short)0, acc[t],
                    /*reuse_a=*/false, /*reuse_b=*/false);
            }
        }
    }

    // Store D: VGPR r -> M = r + 8*lhalf, N = l15
#pragma unroll
    for (int t = 0; t < 4; ++t) {
        const int col = tileN + t * 16 + l15;
#pragma unroll
        for (int r = 0; r < 8; ++r) {
            const int row = tileM + wid * 16 + r + (lhalf << 3);
            Obase[(size_t)row * VV + col] = acc[t][r];
        }
    }
}

// ---------------------------------------------------------------------------
extern "C" void kernel_launch(void* const* d_in, const int* in_sizes, int n_in,
                              void* d_out, int out_size, void* d_ws, size_t ws_size,
                              hipStream_t stream) {
    const float* x      = (const float*)d_in[0];
    const int*   A      = (const int*)d_in[1];
    const float* conv_w = (const float*)d_in[2];
    const float* conv_b = (const float*)d_in[3];
    const float* l2_w   = (const float*)d_in[4];
    const float* l2_b   = (const float*)d_in[5];
    const float* l1_w   = (const float*)d_in[6];
    const float* l1_b   = (const float*)d_in[7];

    float* out = (float*)d_out;                    // (32,64,6,512)
    float* p   = out + OUT_ELEMS;                  // (32,1,512,512), second output

    float* xc  = (float*)d_ws;                     // (32,64,6,512) = 24 MB scratch
    float* e_i = xc + OUT_ELEMS;                   // (32,512)
    float* e_j = e_i + (size_t)NB * VV;            // (32,512)

    conv_pool_e_kernel<<<NB * VV, COUT, 0, stream>>>(x, conv_w, conv_b, l2_w, l2_b,
                                                     l1_w, xc, e_i, e_j);

    attn_softmax_kernel<<<dim3(VV, NB), 256, 0, stream>>>(A, e_i, e_j, l1_b, p);

    gemm_xc_p_kernel<<<dim3(VV / TN, MROWS / TM, NB), 256, 0, stream>>>(xc, p, out);
}